// GIN_2_BN_77558519431975
// MI455X (gfx1250) — compile-verified
//
#include <hip/hip_runtime.h>
#include <math.h>

typedef float v2f __attribute__((ext_vector_type(2)));
typedef float v8f __attribute__((ext_vector_type(8)));

static constexpr int kN   = 50000;   // nodes (multiple of 16)
static constexpr int kDin = 100;     // input features  (25 x K4)
static constexpr int kDh  = 256;     // hidden features (16 n-tiles, 64 x K4)
static constexpr int kDo  = 40;      // output features (3 n-tiles padded to 48)
static constexpr int kMt  = kN / 16; // 3125 row tiles

// ---------------- workspace layout (bytes, all 256B aligned) ----------------
static constexpr size_t AGG1_OFF = 0;          // [N,100] f32 ; region reused below after GEMM1
static constexpr size_t P_OFF    = 0;          // [N,40] f32  (GEMM2 out, reuses agg1 space)
static constexpr size_t AGG2_OFF = 8000000;    // [N,40] f32  (reuses agg1 space)
static constexpr size_t Z_OFF    = 20000000;   // [N,256] f32 (pre-BN layer-1 activations)
static constexpr size_t ST_OFF   = 71200000;   // stats: 1280 floats
// stats float offsets: sum1=0 sumsq1=256 s1=512 t1=768 sum2=1024 q2=1088 s2=1152 t2=1216

// ---------------------------------------------------------------------------
__global__ void zero_stats_k(float* st) {
    for (int i = threadIdx.x; i < 1280; i += blockDim.x) st[i] = 0.0f;
}

__global__ void copy_x_k(const float* __restrict__ x, float* __restrict__ agg1) {
    const int n4 = kN * kDin / 4;
    for (int i = blockIdx.x * blockDim.x + threadIdx.x; i < n4; i += gridDim.x * blockDim.x)
        ((float4*)agg1)[i] = ((const float4*)x)[i];
}

// scatter-add: acc[dst] += feat[src], feature dim D floats (= chunks float4s)
__global__ void edge_scatter_k(const float* __restrict__ feat,
                               const int* __restrict__ esrc, const int* __restrict__ edst,
                               float* __restrict__ acc, int E, int chunks, int D) {
    const int total = E * chunks;
    for (int i = blockIdx.x * blockDim.x + threadIdx.x; i < total; i += gridDim.x * blockDim.x) {
        int e  = i / chunks;
        int ch = i - e * chunks;
        int s  = esrc[e];
        int d  = edst[e];
        float4 v = ((const float4*)(feat + (size_t)s * D))[ch];
        float* a = acc + (size_t)d * D + ch * 4;
        atomicAdd(a + 0, v.x);
        atomicAdd(a + 1, v.y);
        atomicAdd(a + 2, v.z);
        atomicAdd(a + 3, v.w);
    }
}

// z[N,256] = agg1[N,100] @ W1[100,256] + b1 ; one wave per 16x16 tile, f32 WMMA
__global__ void gemm1_wmma_k(const float* __restrict__ agg1, const float* __restrict__ W1,
                             const float* __restrict__ b1, float* __restrict__ z) {
    int wave = (blockIdx.x * blockDim.x + threadIdx.x) >> 5;
    int lane = threadIdx.x & 31;
    int nt = wave & 15;       // 16 n-tiles
    int mt = wave >> 4;
    if (mt >= kMt) return;
    int m0 = mt * 16, n0 = nt * 16;
    int half = lane >> 4, lrow = lane & 15;

    v8f c = {};
    const float* arow = agg1 + (size_t)(m0 + lrow) * kDin;
    const int col = n0 + lrow;
    for (int k0 = 0; k0 < kDin; k0 += 4) {
        int ka = k0 + 2 * half;
        v2f a = *(const v2f*)(arow + ka);                 // A: 16x4, lanes0-15:K0,K1 / lanes16-31:K2,K3
        v2f b;
        b.x = W1[(size_t)ka * kDh + col];                 // B: 4x16, rows striped across lanes
        b.y = W1[(size_t)(ka + 1) * kDh + col];
        c = __builtin_amdgcn_wmma_f32_16x16x4_f32(false, a, false, b, (short)0, c, false, false);
    }
    float bias = b1[col];
#pragma unroll
    for (int v = 0; v < 8; ++v) {
        int row = m0 + v + 8 * half;                      // C/D layout: VGPR v -> M = v (+8 for upper half)
        z[(size_t)row * kDh + col] = c[v] + bias;
    }
}

// per-column sum / sumsq for the 256-wide z (thread t owns column t)
__global__ void colstats256_k(const float* __restrict__ z, float* __restrict__ sum,
                              float* __restrict__ sumsq) {
    int c = threadIdx.x;  // blockDim.x == 256
    float s = 0.f, q = 0.f;
    for (int r = blockIdx.x; r < kN; r += gridDim.x) {
        float v = z[(size_t)r * kDh + c];
        s += v; q += v * v;
    }
    atomicAdd(&sum[c], s);
    atomicAdd(&sumsq[c], q);
}

// scale/shift from sums: s = gamma * rsqrt(var+eps), t = beta - mean*s
__global__ void bn_finalize_k(const float* __restrict__ sum, const float* __restrict__ sumsq,
                              const float* __restrict__ gamma, const float* __restrict__ beta,
                              float* __restrict__ sc, float* __restrict__ sh, int cols) {
    int c = blockIdx.x * blockDim.x + threadIdx.x;
    if (c < cols) {
        const float invN = 1.0f / (float)kN;
        float m   = sum[c] * invN;
        float var = sumsq[c] * invN - m * m;
        float s   = gamma[c] * rsqrtf(var + 1e-5f);
        sc[c] = s;
        sh[c] = beta[c] - m * s;
    }
}

// p[N,40] = relu(z*s1+t1) @ W2[256,40] ; N padded to 48 (3 tiles), fused BN1+ReLU on A
__global__ void gemm2_wmma_k(const float* __restrict__ z, const float* __restrict__ s1,
                             const float* __restrict__ t1, const float* __restrict__ W2,
                             float* __restrict__ p) {
    int wave = (blockIdx.x * blockDim.x + threadIdx.x) >> 5;
    int lane = threadIdx.x & 31;
    int nt = wave % 3;        // 3 n-tiles (cols 0..47, valid < 40)
    int mt = wave / 3;
    if (mt >= kMt) return;
    int m0 = mt * 16, n0 = nt * 16;
    int half = lane >> 4, lrow = lane & 15;
    int col = n0 + lrow;
    bool cok = (col < kDo);

    v8f c = {};
    const float* zrow = z + (size_t)(m0 + lrow) * kDh;
    for (int k0 = 0; k0 < kDh; k0 += 4) {
        int ka = k0 + 2 * half;
        v2f a, b;
        a.x = fmaxf(zrow[ka]     * s1[ka]     + t1[ka],     0.0f);
        a.y = fmaxf(zrow[ka + 1] * s1[ka + 1] + t1[ka + 1], 0.0f);
        b.x = cok ? W2[(size_t)ka * kDo + col]       : 0.0f;
        b.y = cok ? W2[(size_t)(ka + 1) * kDo + col] : 0.0f;
        c = __builtin_amdgcn_wmma_f32_16x16x4_f32(false, a, false, b, (short)0, c, false, false);
    }
    if (cok) {
#pragma unroll
        for (int v = 0; v < 8; ++v) {
            int row = m0 + v + 8 * half;
            p[(size_t)row * kDo + col] = c[v];
        }
    }
}

__global__ void init_agg2_k(const float* __restrict__ p, const float* __restrict__ b2,
                            float* __restrict__ agg2) {
    const int total = kN * kDo;
    for (int i = blockIdx.x * blockDim.x + threadIdx.x; i < total; i += gridDim.x * blockDim.x)
        agg2[i] = p[i] + b2[i % kDo];
}

// per-column sum/sumsq for 40-wide matrix via LDS atomics
__global__ void colstats40_k(const float* __restrict__ a, float* __restrict__ sum,
                             float* __restrict__ sumsq) {
    __shared__ float ls[kDo], lq[kDo];
    if (threadIdx.x < kDo) { ls[threadIdx.x] = 0.f; lq[threadIdx.x] = 0.f; }
    __syncthreads();
    const int total = kN * kDo;
    for (int i = blockIdx.x * blockDim.x + threadIdx.x; i < total; i += gridDim.x * blockDim.x) {
        float v = a[i];
        int c = i % kDo;
        atomicAdd(&ls[c], v);
        atomicAdd(&lq[c], v * v);
    }
    __syncthreads();
    if (threadIdx.x < kDo) {
        atomicAdd(&sum[threadIdx.x],   ls[threadIdx.x]);
        atomicAdd(&sumsq[threadIdx.x], lq[threadIdx.x]);
    }
}

// out[r,:] = log_softmax(agg2[r,:]*s2 + t2) ; one wave per row of 40
__global__ void bn2_logsoftmax_k(const float* __restrict__ agg2, const float* __restrict__ s2,
                                 const float* __restrict__ t2, float* __restrict__ out) {
    int row  = (blockIdx.x * blockDim.x + threadIdx.x) >> 5;
    int lane = threadIdx.x & 31;
    if (row >= kN) return;
    const float* rp = agg2 + (size_t)row * kDo;
    float v0 = rp[lane] * s2[lane] + t2[lane];
    float v1 = (lane < 8) ? (rp[32 + lane] * s2[32 + lane] + t2[32 + lane]) : -3.4e38f;
    float m = fmaxf(v0, v1);
#pragma unroll
    for (int o = 16; o > 0; o >>= 1) m = fmaxf(m, __shfl_xor(m, o, 32));
    float s = expf(v0 - m) + ((lane < 8) ? expf(v1 - m) : 0.0f);
#pragma unroll
    for (int o = 16; o > 0; o >>= 1) s += __shfl_xor(s, o, 32);
    float L = m + logf(s);
    out[(size_t)row * kDo + lane] = v0 - L;
    if (lane < 8) out[(size_t)row * kDo + 32 + lane] = v1 - L;
}

// ---------------------------------------------------------------------------
extern "C" void kernel_launch(void* const* d_in, const int* in_sizes, int n_in,
                              void* d_out, int out_size, void* d_ws, size_t ws_size,
                              hipStream_t stream) {
    const float* x   = (const float*)d_in[0];
    const int*   ei  = (const int*)  d_in[1];
    const float* W1  = (const float*)d_in[2];
    const float* b1  = (const float*)d_in[3];
    const float* W2  = (const float*)d_in[4];
    const float* b2  = (const float*)d_in[5];
    const float* g1  = (const float*)d_in[6];
    const float* be1 = (const float*)d_in[7];
    const float* g2  = (const float*)d_in[8];
    const float* be2 = (const float*)d_in[9];
    const int E = in_sizes[1] / 2;
    const int* esrc = ei;
    const int* edst = ei + E;

    char*  ws   = (char*)d_ws;
    float* agg1 = (float*)(ws + AGG1_OFF);
    float* z    = (float*)(ws + Z_OFF);
    float* p    = (float*)(ws + P_OFF);
    float* agg2 = (float*)(ws + AGG2_OFF);
    float* st   = (float*)(ws + ST_OFF);
    float *sum1 = st, *q1 = st + 256, *s1 = st + 512, *t1 = st + 768;
    float *sum2 = st + 1024, *q2 = st + 1088, *s2 = st + 1152, *t2 = st + 1216;
    float* out  = (float*)d_out;

    // 0) zero BN accumulators (buffers are poisoned / left dirty between replays)
    zero_stats_k<<<1, 256, 0, stream>>>(st);

    // 1) agg1 = x ; agg1[dst] += x[src]
    copy_x_k<<<2048, 256, 0, stream>>>(x, agg1);
    edge_scatter_k<<<4096, 256, 0, stream>>>(x, esrc, edst, agg1, E, kDin / 4, kDin);

    // 2) z = agg1 @ W1 + b1   (f32 WMMA)
    gemm1_wmma_k<<<(kMt * 16) / 8, 256, 0, stream>>>(agg1, W1, b1, z);

    // 3) BN1 stats -> (s1, t1)
    colstats256_k<<<512, 256, 0, stream>>>(z, sum1, q1);
    bn_finalize_k<<<1, 256, 0, stream>>>(sum1, q1, g1, be1, s1, t1, kDh);

    // 4) p = relu(BN1(z)) @ W2   (f32 WMMA, BN+ReLU fused into A loads)
    gemm2_wmma_k<<<(kMt * 3 + 7) / 8, 256, 0, stream>>>(z, s1, t1, W2, p);

    // 5) agg2 = p + b2 ; agg2[dst] += p[src]   (aggregation commuted past the linear: 40-dim)
    init_agg2_k<<<2048, 256, 0, stream>>>(p, b2, agg2);
    edge_scatter_k<<<2048, 256, 0, stream>>>(p, esrc, edst, agg2, E, kDo / 4, kDo);

    // 6) BN2 stats -> (s2, t2)
    colstats40_k<<<1024, 256, 0, stream>>>(agg2, sum2, q2);
    bn_finalize_k<<<1, 64, 0, stream>>>(sum2, q2, g2, be2, s2, t2, kDo);

    // 7) out = log_softmax(BN2(agg2))
    bn2_logsoftmax_k<<<(kN * 32 + 255) / 256, 256, 0, stream>>>(agg2, s2, t2, out);
}